// CiC1D_s_8564164788350
// MI455X (gfx1250) — compile-verified
//
#include <hip/hip_runtime.h>
#include <hip/hip_bf16.h>

// ---------------------------------------------------------------------------
// MI455X (gfx1250) implementation of the CiC1D-style network.
//
// All matmuls (3 convs as implicit GEMM, 3 channel mixes, FC1) run on
// V_WMMA_F32_16X16X4_F32 (fp32-exact matrix pipe). Activations are NHWC
// (c innermost, 64-wide): A fragments are 8B v2f loads, stores lane-coalesced.
// Weights are pre-transposed to K-major *paired* layout [k/2][n][2] so each
// B fragment is a single 8B v2f load. The big convs use 2x2 register blocking
// (wave = 32 pixels x 32 channels, 4 accumulators) -> 1 vmem per WMMA.
// M = 128*OH*OW is always divisible by 64, so all tiles are full (no guards,
// EXEC stays all-ones as WMMA requires). FC1 is split-K (16 chunks) with a
// deterministic reduce. BN stats: LDS block reduction + f32 global atomics.
// ---------------------------------------------------------------------------

typedef float v2f __attribute__((ext_vector_type(2)));
typedef float v8f __attribute__((ext_vector_type(8)));

#define EPSV 1e-5f

__device__ __forceinline__ v8f wmma4(v2f a, v2f b, v8f c) {
  // D(16x16,f32) = A(16x4,f32) * B(4x16,f32) + C
  return __builtin_amdgcn_wmma_f32_16x16x4_f32(false, a, false, b, (short)0, c,
                                               false, false);
}

// ------------------------------- utility -----------------------------------

__global__ void k_zero(float* __restrict__ p, int n) {
  int i = blockIdx.x * blockDim.x + threadIdx.x;
  if (i < n) p[i] = 0.f;
}

// W (64,64,KH,KW) OIHW -> paired K-major: wt[(k>>1)*128 + n*2 + (k&1)],
// k = p*64 + ci. B fragment (B[k][n],B[k+1][n]) is then one v2f load.
__global__ void k_tw64p(const float* __restrict__ w, float* __restrict__ wt, int Pk) {
  int idx = blockIdx.x * blockDim.x + threadIdx.x;
  int total = Pk * 64 * 64;
  if (idx >= total) return;
  int e  = idx & 1;
  int n  = (idx >> 1) & 63;
  int kp = idx >> 7;
  int k  = kp * 2 + e;
  int ci = k & 63;
  int p  = k >> 6;
  wt[idx] = w[((size_t)n * 64 + ci) * Pk + p];
}

// W11 (64,1,20,8) -> paired: wt[(p>>1)*128 + n*2 + (p&1)]
__global__ void k_tw1p(const float* __restrict__ w, float* __restrict__ wt, int Pk) {
  int idx = blockIdx.x * blockDim.x + threadIdx.x;
  if (idx >= Pk * 64) return;
  int e  = idx & 1;
  int n  = (idx >> 1) & 63;
  int p  = (idx >> 7) * 2 + e;
  wt[idx] = w[n * Pk + p];
}

// Wm (64,62) -> paired [d/2][n][2], rows d=62,63 zero (kills K padding).
__global__ void k_twmixp(const float* __restrict__ w, float* __restrict__ wt) {
  int idx = blockIdx.x * blockDim.x + threadIdx.x;
  if (idx >= 64 * 64) return;
  int e = idx & 1;
  int n = (idx >> 1) & 63;
  int d = (idx >> 7) * 2 + e;
  wt[idx] = (d < 62) ? w[n * 62 + d] : 0.f;
}

// Wf1 (64,68096), ref flatten k = c*1064 + s (NCHW). Our NHWC flatten is
// k' = s*64 + c. Store paired K-major on k'.
__global__ void k_twfcp(const float* __restrict__ w, float* __restrict__ wt) {
  int idx = blockIdx.x * blockDim.x + threadIdx.x;
  if (idx >= 64 * 68096) return;
  int n = idx / 68096;
  int k = idx - n * 68096;
  int c = k / 1064;
  int s = k - c * 1064;
  int kp = s * 64 + c;
  wt[(size_t)(kp >> 1) * 128 + n * 2 + (kp & 1)] = w[idx];
}

// ------------------------------ conv kernels --------------------------------

// conv1: x (128,1,98,40) -> out NHWC (P1,64). K = 160 taps. 1x1 tile/wave.
__global__ void k_conv1(const float* __restrict__ x, const float* __restrict__ wt,
                        const float* __restrict__ bias, float* __restrict__ out) {
  const int IH = 98, IW = 40, OH = 79, OW = 33, KK = 160;
  int lane = threadIdx.x & 31, wave = threadIdx.x >> 5;
  int hi = lane >> 4, ln = lane & 15;
  int tile = blockIdx.x;
  int m = tile * 16 + ln;            // M % 16 == 0: tile always full
  int b = m / (OH * OW);
  int r = m - b * (OH * OW);
  int oy = r / OW, ox = r - (r / OW) * OW;
  const float* xin = x + ((size_t)b * IH + oy) * IW + ox;
  int n = wave * 16 + ln;
  v8f acc = {};
  for (int kc = 0; kc < KK; kc += 4) {
    int ka = kc + 2 * hi;
    v2f a;
    a.x = xin[(ka >> 3) * IW + (ka & 7)];
    a.y = xin[((ka + 1) >> 3) * IW + ((ka + 1) & 7)];
    v2f bb = *(const v2f*)(wt + (ka >> 1) * 128 + n * 2);
    acc = wmma4(a, bb, acc);
  }
  float bi = bias[n];
#pragma unroll
  for (int i = 0; i < 8; ++i)
    out[(size_t)(tile * 16 + 8 * hi + i) * 64 + n] = acc[i] + bi;
}

// 64->64 conv, NHWC in/out, paired K-major weights. 2x2 register blocking:
// wave = 32 pixels x 32 channels; block = 4 waves = 64 pixels x 64 channels.
__global__ void k_conv64(const float* __restrict__ in, const float* __restrict__ wt,
                         const float* __restrict__ bias, float* __restrict__ out,
                         int IH, int IW, int OH, int OW, int KH, int KW) {
  int lane = threadIdx.x & 31, wave = threadIdx.x >> 5;
  int hi = lane >> 4, ln = lane & 15;
  int waveM = wave >> 1, waveN = wave & 1;
  int OHOW = OH * OW;
  int base = blockIdx.x * 64 + waveM * 32;     // M % 64 == 0 always
  int m0 = base + ln, m1 = base + 16 + ln;
  int b0 = m0 / OHOW, r0 = m0 - b0 * OHOW;
  int oy0 = r0 / OW, ox0 = r0 - oy0 * OW;
  int b1 = m1 / OHOW, r1 = m1 - b1 * OHOW;
  int oy1 = r1 / OW, ox1 = r1 - oy1 * OW;
  int n0 = waveN * 32 + ln, n1 = n0 + 16;
  v8f acc00 = {}, acc01 = {}, acc10 = {}, acc11 = {};
  for (int ky = 0; ky < KH; ++ky) {
    const float* arow0 = in + (((size_t)b0 * IH + oy0 + ky) * IW + ox0) * 64;
    const float* arow1 = in + (((size_t)b1 * IH + oy1 + ky) * IW + ox1) * 64;
    const float* wrow  = wt + (size_t)(ky * KW) * 4096;
    for (int kx = 0; kx < KW; ++kx) {
      const float* ap0 = arow0 + kx * 64;
      const float* ap1 = arow1 + kx * 64;
      const float* wp  = wrow + (size_t)kx * 4096;
#pragma unroll
      for (int ci = 0; ci < 64; ci += 4) {
        int ka = ci + 2 * hi;
        int kp = (ci >> 1) + hi;             // (ka>>1), ci even
        v2f a0 = *(const v2f*)(ap0 + ka);
        v2f a1 = *(const v2f*)(ap1 + ka);
        v2f bb0 = *(const v2f*)(wp + kp * 128 + n0 * 2);
        v2f bb1 = *(const v2f*)(wp + kp * 128 + n1 * 2);
        acc00 = wmma4(a0, bb0, acc00);
        acc01 = wmma4(a0, bb1, acc01);
        acc10 = wmma4(a1, bb0, acc10);
        acc11 = wmma4(a1, bb1, acc11);
      }
    }
  }
  float bi0 = bias[n0], bi1 = bias[n1];
#pragma unroll
  for (int i = 0; i < 8; ++i) {
    size_t mr0 = (size_t)(base + 8 * hi + i) * 64;
    size_t mr1 = (size_t)(base + 16 + 8 * hi + i) * 64;
    out[mr0 + n0] = acc00[i] + bi0;
    out[mr0 + n1] = acc01[i] + bi1;
    out[mr1 + n0] = acc10[i] + bi0;
    out[mr1 + n1] = acc11[i] + bi1;
  }
}

// --------------------------- BN stats / apply -------------------------------

// per-channel sum/sumsq over NHWC tensor (total = P*64). gridDim*blockDim is a
// multiple of 64 so each thread's channel is fixed.
__global__ void k_stats_ch(const float* __restrict__ t, float* __restrict__ sum,
                           float* __restrict__ sq, size_t total) {
  size_t stride = (size_t)gridDim.x * blockDim.x;
  size_t i0 = (size_t)blockIdx.x * blockDim.x + threadIdx.x;
  float s = 0.f, q = 0.f;
  for (size_t i = i0; i < total; i += stride) {
    float v = t[i];
    s += v;
    q += v * v;
  }
  __shared__ float ls[256], lq[256];
  ls[threadIdx.x] = s;
  lq[threadIdx.x] = q;
  __syncthreads();
  if (threadIdx.x < 64) {
    float S = ls[threadIdx.x] + ls[threadIdx.x + 64] + ls[threadIdx.x + 128] +
              ls[threadIdx.x + 192];
    float Q = lq[threadIdx.x] + lq[threadIdx.x + 64] + lq[threadIdx.x + 128] +
              lq[threadIdx.x + 192];
    atomicAdd(&sum[threadIdx.x], S);
    atomicAdd(&sq[threadIdx.x], Q);
  }
}

__global__ void k_apply_ch(float* __restrict__ t, const float* __restrict__ sum,
                           const float* __restrict__ sq, const float* __restrict__ g,
                           const float* __restrict__ be, float invN, size_t total,
                           int do_relu) {
  size_t stride = (size_t)gridDim.x * blockDim.x;
  size_t i = (size_t)blockIdx.x * blockDim.x + threadIdx.x;
  int c = (int)(i & 63);
  float m  = sum[c] * invN;
  float vr = sq[c] * invN - m * m;
  float sc = g[c] * rsqrtf(vr + EPSV);
  float sh = be[c] - m * sc;
  for (; i < total; i += stride) {
    float v = t[i] * sc + sh;
    t[i] = do_relu ? fmaxf(v, 0.f) : v;
  }
}

// ------------------------------- LC block -----------------------------------

// t[q,c] = lcb[c] + sum_k h[q,c+k]*lcw[c,k]  (c<62; pad channels -> 0), plus
// global sum/sumsq over the 62 real channels for _bn_all.
__global__ void k_lc(const float* __restrict__ h, float* __restrict__ t,
                     const float* __restrict__ lcw, const float* __restrict__ lcb,
                     float* __restrict__ allstats, size_t total) {
  size_t stride = (size_t)gridDim.x * blockDim.x;
  float s = 0.f, q = 0.f;
  for (size_t i = (size_t)blockIdx.x * blockDim.x + threadIdx.x; i < total;
       i += stride) {
    int c = (int)(i & 63);
    float v = 0.f;
    if (c < 62) {
      v = lcb[c] + h[i] * lcw[c * 3] + h[i + 1] * lcw[c * 3 + 1] +
          h[i + 2] * lcw[c * 3 + 2];
      s += v;
      q += v * v;
    }
    t[i] = v;
  }
  __shared__ float ls[256], lq[256];
  ls[threadIdx.x] = s;
  lq[threadIdx.x] = q;
  __syncthreads();
  for (int o = 128; o > 0; o >>= 1) {
    if (threadIdx.x < o) {
      ls[threadIdx.x] += ls[threadIdx.x + o];
      lq[threadIdx.x] += lq[threadIdx.x + o];
    }
    __syncthreads();
  }
  if (threadIdx.x == 0) {
    atomicAdd(&allstats[0], ls[0]);
    atomicAdd(&allstats[1], lq[0]);
  }
}

// channel mix: u[q,n] = sum_d relu(bn_all(t[q,d])) * Wm[n,d] + bias[n].
// GEMM M=P (P%16==0), K=64 (padded; B rows 62/63 zero), N=64.
__global__ void k_mix(const float* __restrict__ t, const float* __restrict__ wt,
                      const float* __restrict__ bias,
                      const float* __restrict__ allstats,
                      const float* __restrict__ g1, const float* __restrict__ b1,
                      float invCount, float* __restrict__ u) {
  int lane = threadIdx.x & 31, wave = threadIdx.x >> 5;
  int hi = lane >> 4, ln = lane & 15;
  int tile = blockIdx.x;
  int m = tile * 16 + ln;
  float mean = allstats[0] * invCount;
  float var  = allstats[1] * invCount - mean * mean;
  float sc   = g1[0] * rsqrtf(var + EPSV);
  float sh   = b1[0] - mean * sc;
  const float* trow = t + (size_t)m * 64;
  int n = wave * 16 + ln;
  v8f acc = {};
#pragma unroll
  for (int kc = 0; kc < 64; kc += 4) {
    int ka = kc + 2 * hi;
    v2f raw = *(const v2f*)(trow + ka);
    v2f a;
    a.x = fmaxf(raw.x * sc + sh, 0.f);
    a.y = fmaxf(raw.y * sc + sh, 0.f);
    v2f bb = *(const v2f*)(wt + ((kc >> 1) + hi) * 128 + n * 2);
    acc = wmma4(a, bb, acc);
  }
  float bi = bias[n];
#pragma unroll
  for (int i = 0; i < 8; ++i)
    u[(size_t)(tile * 16 + 8 * hi + i) * 64 + n] = acc[i] + bi;
}

// ------------------------------- FC + head ----------------------------------

// FC1 split-K: grid = (8 M-tiles, NCHUNK). Each block accumulates its K chunk
// into partial[chunk][128][64]; reduce adds chunks + bias.
#define FC_K 68096
#define FC_NCHUNK 16
#define FC_KC (FC_K / FC_NCHUNK)   // 4256, divisible by 4

__global__ void k_fc1(const float* __restrict__ h, const float* __restrict__ wt,
                      float* __restrict__ partial) {
  int lane = threadIdx.x & 31, wave = threadIdx.x >> 5;
  int hi = lane >> 4, ln = lane & 15;
  int tile = blockIdx.x;                 // 0..7
  int chunk = blockIdx.y;                // 0..15
  int m = tile * 16 + ln;
  int n = wave * 16 + ln;
  const float* hrow = h + (size_t)m * FC_K;
  int k0 = chunk * FC_KC;
  v8f acc = {};
  for (int kc = k0; kc < k0 + FC_KC; kc += 4) {
    int ka = kc + 2 * hi;
    v2f a = *(const v2f*)(hrow + ka);
    v2f bb = *(const v2f*)(wt + (size_t)(ka >> 1) * 128 + n * 2);
    acc = wmma4(a, bb, acc);
  }
  float* po = partial + (size_t)chunk * 8192;
#pragma unroll
  for (int i = 0; i < 8; ++i)
    po[(tile * 16 + 8 * hi + i) * 64 + n] = acc[i];
}

__global__ void k_fc1red(const float* __restrict__ partial,
                         const float* __restrict__ bias, float* __restrict__ outf) {
  int i = blockIdx.x * blockDim.x + threadIdx.x;
  if (i >= 8192) return;
  float s = bias[i & 63];
#pragma unroll
  for (int c = 0; c < FC_NCHUNK; ++c) s += partial[c * 8192 + i];
  outf[i] = s;
}

// batch-BN (per feature over 128 rows) + relu + (128x64)@(64x35)^T + bias
__global__ void k_head(const float* __restrict__ f, const float* __restrict__ gf,
                       const float* __restrict__ bef, const float* __restrict__ w2,
                       const float* __restrict__ b2, float* __restrict__ out) {
  __shared__ float hn[128 * 64];
  __shared__ float sc[64], sh[64];
  int tid = threadIdx.x;                 // block = 256
  if (tid < 64) {
    float s = 0.f, q = 0.f;
    for (int b = 0; b < 128; ++b) {
      float v = f[b * 64 + tid];
      s += v;
      q += v * v;
    }
    float m  = s * (1.f / 128.f);
    float vr = q * (1.f / 128.f) - m * m;
    float scale = gf[tid] * rsqrtf(vr + EPSV);
    sc[tid] = scale;
    sh[tid] = bef[tid] - m * scale;
  }
  __syncthreads();
  for (int i = tid; i < 128 * 64; i += 256) {
    int c = i & 63;
    hn[i] = fmaxf(f[i] * sc[c] + sh[c], 0.f);
  }
  __syncthreads();
  for (int i = tid; i < 128 * 35; i += 256) {
    int b = i / 35, o = i - b * 35;
    float s = b2[o];
    const float* hb = &hn[b * 64];
    const float* wo = &w2[o * 64];
#pragma unroll
    for (int c = 0; c < 64; ++c) s += hb[c] * wo[c];
    out[i] = s;
  }
}

// ------------------------------- launcher -----------------------------------

extern "C" void kernel_launch(void* const* d_in, const int* in_sizes, int n_in,
                              void* d_out, int out_size, void* d_ws, size_t ws_size,
                              hipStream_t stream) {
  (void)in_sizes; (void)n_in; (void)out_size; (void)ws_size;
  const float* x    = (const float*)d_in[0];
  const float* W11  = (const float*)d_in[1];
  const float* b11  = (const float*)d_in[2];
  const float* g11  = (const float*)d_in[3];
  const float* be11 = (const float*)d_in[4];
  const float* lc1w = (const float*)d_in[5];
  const float* lc1b = (const float*)d_in[6];
  const float* g12  = (const float*)d_in[7];
  const float* be12 = (const float*)d_in[8];
  const float* W13  = (const float*)d_in[9];
  const float* b13  = (const float*)d_in[10];
  const float* g13  = (const float*)d_in[11];
  const float* be13 = (const float*)d_in[12];
  const float* W21  = (const float*)d_in[13];
  const float* b21  = (const float*)d_in[14];
  const float* g21  = (const float*)d_in[15];
  const float* be21 = (const float*)d_in[16];
  const float* lc2w = (const float*)d_in[17];
  const float* lc2b = (const float*)d_in[18];
  const float* g22  = (const float*)d_in[19];
  const float* be22 = (const float*)d_in[20];
  const float* W23  = (const float*)d_in[21];
  const float* b23  = (const float*)d_in[22];
  const float* g23  = (const float*)d_in[23];
  const float* be23 = (const float*)d_in[24];
  const float* W31  = (const float*)d_in[25];
  const float* b31  = (const float*)d_in[26];
  const float* g31  = (const float*)d_in[27];
  const float* be31 = (const float*)d_in[28];
  const float* lc3w = (const float*)d_in[29];
  const float* lc3b = (const float*)d_in[30];
  const float* g32  = (const float*)d_in[31];
  const float* be32 = (const float*)d_in[32];
  const float* W33  = (const float*)d_in[33];
  const float* b33  = (const float*)d_in[34];
  const float* g33  = (const float*)d_in[35];
  const float* be33 = (const float*)d_in[36];
  const float* Wf1  = (const float*)d_in[37];
  const float* bf1  = (const float*)d_in[38];
  const float* gf   = (const float*)d_in[39];
  const float* bef  = (const float*)d_in[40];
  const float* Wf2  = (const float*)d_in[41];
  const float* bf2  = (const float*)d_in[42];
  float* out = (float*)d_out;
  float* W = (float*)d_ws;

  // geometry (all P divisible by 64)
  const int P1 = 128 * 79 * 33;   // 333696
  const int P2 = 128 * 65 * 26;   // 216320
  const int P3 = 128 * 56 * 19;   // 136192
  const size_t BIG = (size_t)P1 * 64;

  // workspace layout (floats)
  float* chsum  = W + 0;            // 64
  float* chsq   = W + 64;           // 64
  float* allst  = W + 128;          // 2
  float* fc1o   = W + 256;          // 8192
  float* fc1par = W + 8448;         // 16*8192 = 131072
  float* wt1    = W + 139520;       // 160*64
  float* wt2    = W + 149760;       // 7680*64
  float* wt3    = W + 641280;       // 5120*64
  float* wmix   = W + 968960;       // 64*64
  float* wfc    = W + 973056;       // 68096*64
  float* bufA   = W + 5331200;      // BIG
  float* bufB   = bufA + BIG;
  float* bufC   = bufB + BIG;

  auto cdiv = [](long long a, long long b) { return (int)((a + b - 1) / b); };
  const int RED_GRID = 2048;

  // ---- weight transforms (once per launch; tiny vs conv cost) ----
  k_tw1p <<<cdiv(160 * 64, 256), 256, 0, stream>>>(W11, wt1, 160);
  k_tw64p<<<cdiv(120 * 64 * 64, 256), 256, 0, stream>>>(W21, wt2, 120);
  k_tw64p<<<cdiv(80 * 64 * 64, 256), 256, 0, stream>>>(W31, wt3, 80);
  k_twfcp<<<cdiv(64 * 68096, 256), 256, 0, stream>>>(Wf1, wfc);

  // ---- stage 1 ----
  k_conv1<<<P1 / 16, 128, 0, stream>>>(x, wt1, b11, bufA);
  k_zero<<<1, 128, 0, stream>>>(chsum, 128);
  k_stats_ch<<<RED_GRID, 256, 0, stream>>>(bufA, chsum, chsq, (size_t)P1 * 64);
  k_apply_ch<<<RED_GRID, 256, 0, stream>>>(bufA, chsum, chsq, g11, be11,
                                           1.f / P1, (size_t)P1 * 64, 1);
  k_zero<<<1, 32, 0, stream>>>(allst, 2);
  k_lc<<<RED_GRID, 256, 0, stream>>>(bufA, bufB, lc1w, lc1b, allst, (size_t)P1 * 64);
  k_twmixp<<<16, 256, 0, stream>>>(W13, wmix);
  k_mix<<<P1 / 16, 128, 0, stream>>>(bufB, wmix, b13, allst, g12, be12,
                                     1.f / ((float)P1 * 62.f), bufC);
  k_zero<<<1, 128, 0, stream>>>(chsum, 128);
  k_stats_ch<<<RED_GRID, 256, 0, stream>>>(bufC, chsum, chsq, (size_t)P1 * 64);
  k_apply_ch<<<RED_GRID, 256, 0, stream>>>(bufC, chsum, chsq, g13, be13,
                                           1.f / P1, (size_t)P1 * 64, 0);

  // ---- stage 2 ----
  k_conv64<<<P2 / 64, 128, 0, stream>>>(bufC, wt2, b21, bufA, 79, 33, 65, 26, 15, 8);
  k_zero<<<1, 128, 0, stream>>>(chsum, 128);
  k_stats_ch<<<RED_GRID, 256, 0, stream>>>(bufA, chsum, chsq, (size_t)P2 * 64);
  k_apply_ch<<<RED_GRID, 256, 0, stream>>>(bufA, chsum, chsq, g21, be21,
                                           1.f / P2, (size_t)P2 * 64, 1);
  k_zero<<<1, 32, 0, stream>>>(allst, 2);
  k_lc<<<RED_GRID, 256, 0, stream>>>(bufA, bufB, lc2w, lc2b, allst, (size_t)P2 * 64);
  k_twmixp<<<16, 256, 0, stream>>>(W23, wmix);
  k_mix<<<P2 / 16, 128, 0, stream>>>(bufB, wmix, b23, allst, g22, be22,
                                     1.f / ((float)P2 * 62.f), bufC);
  k_zero<<<1, 128, 0, stream>>>(chsum, 128);
  k_stats_ch<<<RED_GRID, 256, 0, stream>>>(bufC, chsum, chsq, (size_t)P2 * 64);
  k_apply_ch<<<RED_GRID, 256, 0, stream>>>(bufC, chsum, chsq, g23, be23,
                                           1.f / P2, (size_t)P2 * 64, 0);

  // ---- stage 3 ----
  k_conv64<<<P3 / 64, 128, 0, stream>>>(bufC, wt3, b31, bufA, 65, 26, 56, 19, 10, 8);
  k_zero<<<1, 128, 0, stream>>>(chsum, 128);
  k_stats_ch<<<RED_GRID, 256, 0, stream>>>(bufA, chsum, chsq, (size_t)P3 * 64);
  k_apply_ch<<<RED_GRID, 256, 0, stream>>>(bufA, chsum, chsq, g31, be31,
                                           1.f / P3, (size_t)P3 * 64, 1);
  k_zero<<<1, 32, 0, stream>>>(allst, 2);
  k_lc<<<RED_GRID, 256, 0, stream>>>(bufA, bufB, lc3w, lc3b, allst, (size_t)P3 * 64);
  k_twmixp<<<16, 256, 0, stream>>>(W33, wmix);
  k_mix<<<P3 / 16, 128, 0, stream>>>(bufB, wmix, b33, allst, g32, be32,
                                     1.f / ((float)P3 * 62.f), bufC);
  k_zero<<<1, 128, 0, stream>>>(chsum, 128);
  k_stats_ch<<<RED_GRID, 256, 0, stream>>>(bufC, chsum, chsq, (size_t)P3 * 64);
  k_apply_ch<<<RED_GRID, 256, 0, stream>>>(bufC, chsum, chsq, g33, be33,
                                           1.f / P3, (size_t)P3 * 64, 0);

  // ---- FC + head ----
  {
    dim3 g(8, FC_NCHUNK);
    k_fc1<<<g, 128, 0, stream>>>(bufC, wfc, fc1par);
  }
  k_fc1red<<<32, 256, 0, stream>>>(fc1par, bf1, fc1o);
  k_head<<<1, 256, 0, stream>>>(fc1o, gf, bef, Wf2, bf2, out);
}